// ROIAlign_47639777247768
// MI455X (gfx1250) — compile-verified
//
#include <hip/hip_runtime.h>

typedef float v2f __attribute__((ext_vector_type(2)));
typedef float v8f __attribute__((ext_vector_type(8)));

#define OUTX 7           // pooled output is 7x7
#define SS   14          // sampling grid is 14x14 (OUT*NS)
#define CBLK 16          // channels per wave (WMMA N dim)
#define WPW  8           // waves per workgroup
#define LDSF (4 * SS * CBLK)   // per-wave LDS floats: 4 k-rows x 14 cols x 16 ch

__global__ __launch_bounds__(WPW * 32)
void roialign_cdna5_wmma(const float* __restrict__ f0,
                         const float* __restrict__ f1,
                         const float* __restrict__ f2,
                         const float* __restrict__ f3,
                         const float* __restrict__ props,
                         float* __restrict__ out,
                         int nroi, int rois_per_img)
{
    __shared__ float lds[WPW * LDSF];   // 28 KB

    const int lane = threadIdx.x & 31;
    const int wave = threadIdx.x >> 5;
    const int li   = lane & 15;         // WMMA M-row (x-stage) / N-col channel
    const int grp  = lane >> 4;         // lane group selects K half

    // pair id is wave-uniform: force it scalar so control flow stays uniform
    const int pair = __builtin_amdgcn_readfirstlane((int)blockIdx.x * WPW + wave);
    const int roi  = pair >> 4;
    if (roi >= nroi) return;
    const int c0  = (pair & 15) * CBLK;
    const int img = roi / rois_per_img;

    const float bx1 = props[roi * 4 + 0];
    const float by1 = props[roi * 4 + 1];
    const float bx2 = props[roi * 4 + 2];
    const float by2 = props[roi * 4 + 3];

    // ---- FPN level selection (matches reference _levels) ----
    const float area = (bx2 - bx1) * (by2 - by1);
    int lvl = (int)floorf(2.0f + log2f(sqrtf(area) * (1.0f / 224.0f)));
    lvl = lvl < 0 ? 0 : (lvl > 3 ? 3 : lvl);
    lvl = __builtin_amdgcn_readfirstlane(lvl);
    const float scale = (lvl == 0) ? 0.25f : (lvl == 1) ? 0.125f
                       : (lvl == 2) ? 0.0625f : 0.03125f;
    const int Wd = __builtin_amdgcn_readfirstlane(256 >> lvl);   // H == W
    const float* feat = (lvl == 0) ? f0 : (lvl == 1) ? f1 : (lvl == 2) ? f2 : f3;

    const float sx1 = bx1 * scale, sy1 = by1 * scale;
    const float sx2 = bx2 * scale, sy2 = by2 * scale;
    const float wu = (sx2 - sx1) * (1.0f / SS);
    const float hu = (sy2 - sy1) * (1.0f / SS);
    const float fWm1 = (float)(Wd - 1);

    // ---- per-lane x sampling column params (rows of Wx for i = li) ----
    float xi = sx1 + ((float)li + 0.5f) * wu;
    int xf = (int)floorf(xi);
    int x0c = xf < 0 ? 0 : (xf > Wd - 1 ? Wd - 1 : xf);
    int x1c = (xf + 1) > Wd - 1 ? Wd - 1 : (xf + 1);
    float xcc = fminf(fmaxf(xi, 0.0f), fWm1);
    float wxa = (float)x1c - xcc;        // weight at clamped col x0c
    float wxb = xcc - (float)x0c;        // weight at clamped col x1c
    x0c = (li < SS) ? x0c : -0x40000000; // pad rows of Wx are all-zero
    x1c = (li < SS) ? x1c : -0x40000000;

    // ---- per-lane y sampling row params (rows of Wy for j = li) ----
    float yj = sy1 + ((float)li + 0.5f) * hu;
    int yf = (int)floorf(yj);
    int y0c = yf < 0 ? 0 : (yf > Wd - 1 ? Wd - 1 : yf);
    int y1c = (yf + 1) > Wd - 1 ? Wd - 1 : (yf + 1);
    float ycc = fminf(fmaxf(yj, 0.0f), fWm1);
    float wya = (float)y1c - ycc;
    float wyb = ycc - (float)y0c;
    y0c = (li < SS) ? y0c : -0x40000000;
    y1c = (li < SS) ? y1c : -0x40000000;

    // ---- wave-uniform patch bounds (grid coords are monotone in i/j) ----
    int px0 = (int)floorf(sx1 + 0.5f * wu);
    px0 = px0 < 0 ? 0 : (px0 > Wd - 1 ? Wd - 1 : px0);
    int px1 = (int)floorf(sx1 + ((float)SS - 0.5f) * wu) + 1;
    px1 = px1 < 0 ? 0 : (px1 > Wd - 1 ? Wd - 1 : px1);
    int py0 = (int)floorf(sy1 + 0.5f * hu);
    py0 = py0 < 0 ? 0 : (py0 > Wd - 1 ? Wd - 1 : py0);
    int py1 = (int)floorf(sy1 + ((float)SS - 0.5f) * hu) + 1;
    py1 = py1 < 0 ? 0 : (py1 > Wd - 1 ? Wd - 1 : py1);
    px0 = __builtin_amdgcn_readfirstlane(px0);
    px1 = __builtin_amdgcn_readfirstlane(px1);
    py0 = __builtin_amdgcn_readfirstlane(py0);
    py1 = __builtin_amdgcn_readfirstlane(py1);
    const int Wp4 = __builtin_amdgcn_readfirstlane((px1 - px0 + 1 + 3) & ~3);
    const int Hp  = __builtin_amdgcn_readfirstlane(py1 - py0 + 1);
    const int Hp4 = (Hp + 3) & ~3;

    // this lane's channel plane (B-operand N index = li)
    const float* fchan = feat +
        ((size_t)img * 256 + (size_t)(c0 + li)) * (size_t)Wd * (size_t)Wd;
    float* myLds = lds + wave * LDSF;

    // 14 y-stage accumulator tiles: E[i][q] = out[j = q + 8*grp][i][c = li]
    v8f E[SS];
#pragma unroll
    for (int i = 0; i < SS; ++i) E[i] = (v8f){0, 0, 0, 0, 0, 0, 0, 0};

    for (int r4 = 0; r4 < Hp4; r4 += 4) {
        // prefetch next row chunk (global_prefetch_b8)
        {
            int rn = r4 + 4; rn = rn < Hp ? rn : Hp - 1;
            __builtin_prefetch(fchan + (size_t)(py0 + rn) * Wd + px0, 0, 0);
        }

        // ---- x-interp: Rx_r(14x16) = Wx(14xWp) * Patch_r(Wpx16), K-chunks of 4
#pragma unroll
        for (int sub = 0; sub < 4; ++sub) {
            int rIdx = r4 + sub;
            int rcl = rIdx < Hp ? rIdx : Hp - 1;    // clamp pad rows (weight=0 later)
            const float* rowp = fchan + (size_t)(py0 + rcl) * Wd;
            v8f D = (v8f){0, 0, 0, 0, 0, 0, 0, 0};
            for (int kx = 0; kx < Wp4; kx += 4) {
                // K index mapping: k = vgpr + 2*lanegroup
                int colk0 = px0 + kx + (grp << 1);
                int colk1 = colk0 + 1;
                int cc0 = colk0 < px1 ? colk0 : px1;  // safe address for pad cols
                int cc1 = colk1 < px1 ? colk1 : px1;
                v2f Bop;
                Bop.x = rowp[cc0];
                Bop.y = rowp[cc1];
                v2f Aop;   // Wx[li][colk] = 2-nonzero sparse row, built on the fly
                Aop.x = (colk0 == x0c ? wxa : 0.0f) + (colk0 == x1c ? wxb : 0.0f);
                Aop.y = (colk1 == x0c ? wxa : 0.0f) + (colk1 == x1c ? wxb : 0.0f);
                D = __builtin_amdgcn_wmma_f32_16x16x4_f32(
                        false, Aop, false, Bop, (short)0, D, false, false);
            }
            // stage D (C-layout: M = q + 8*grp) into LDS as Rx[sub][i][c]
#pragma unroll
            for (int q = 0; q < 8; ++q) {
                int irow = q + 8 * grp;
                if (irow < SS)
                    myLds[(sub * SS + irow) * CBLK + li] = D[q];
            }
        }

        // ---- y-interp accumulate: E_i += Wy(14x4) * Rx[., i, .](4x16) ----
        const int s0  = grp << 1;            // K index = vgpr + 2*lanegroup
        const int rk0 = py0 + r4 + s0;       // global feature row for this K slot
#pragma unroll
        for (int i = 0; i < SS; ++i) {
            v2f Bop;
            Bop.x = myLds[(s0 * SS + i) * CBLK + li];
            Bop.y = myLds[((s0 + 1) * SS + i) * CBLK + li];
            v2f Aop;   // Wy[li][row]; pad rows (row > py1) get weight 0
            Aop.x = (rk0 == y0c ? wya : 0.0f) + (rk0 == y1c ? wyb : 0.0f);
            Aop.y = ((rk0 + 1) == y0c ? wya : 0.0f) + ((rk0 + 1) == y1c ? wyb : 0.0f);
            E[i] = __builtin_amdgcn_wmma_f32_16x16x4_f32(
                       false, Aop, false, Bop, (short)0, E[i], false, false);
        }
    }

    // ---- 2x2 adaptive max pool (exact for 14 -> 7) and scatter store ----
    const size_t obase = ((size_t)roi * 256 + (size_t)(c0 + li)) * (OUTX * OUTX);
#pragma unroll
    for (int io = 0; io < OUTX; ++io) {
        v8f M;
#pragma unroll
        for (int q = 0; q < 8; ++q) M[q] = fmaxf(E[2 * io][q], E[2 * io + 1][q]);
#pragma unroll
        for (int qp = 0; qp < 4; ++qp) {
            int jo = qp + (grp ? 4 : 0);     // j rows q+8*grp pooled pairwise
            float v = fmaxf(M[2 * qp], M[2 * qp + 1]);
            if (jo < OUTX)
                out[obase + (size_t)(jo * OUTX + io)] = v;
        }
    }
}

extern "C" void kernel_launch(void* const* d_in, const int* in_sizes, int n_in,
                              void* d_out, int out_size, void* d_ws, size_t ws_size,
                              hipStream_t stream) {
    const float* f0 = (const float*)d_in[0];   // (B,256,256,256)
    const float* f1 = (const float*)d_in[1];   // (B,256,128,128)
    const float* f2 = (const float*)d_in[2];   // (B,256,64,64)
    const float* f3 = (const float*)d_in[3];   // (B,256,32,32)
    const float* pr = (const float*)d_in[4];   // (B,512,4)
    float* out = (float*)d_out;                // (B*512, 256, 7, 7)

    const int nroi = in_sizes[4] / 4;
    const int nimg = in_sizes[0] / (256 * 256 * 256);
    const int rois_per_img = nroi / (nimg > 0 ? nimg : 1);

    const int pairs  = nroi * 16;              // (roi, 16-channel block)
    const int blocks = (pairs + WPW - 1) / WPW;
    roialign_cdna5_wmma<<<blocks, WPW * 32, 0, stream>>>(
        f0, f1, f2, f3, pr, out, nroi, rois_per_img);
}